// SetEq3to3_15324443312599
// MI455X (gfx1250) — compile-verified
//
#include <hip/hip_runtime.h>
#include <hip/hip_bf16.h>

// Problem constants (from reference): N=2, D_IN=D_OUT=8, M=64, BASIS=19
#define NN   2
#define DD   8
#define SS   8
#define MM   64
#define M2   4096        // 64*64
#define M3   262144      // 64^3
#define CB   19          // basis dim (coefs innermost stride)
#define CD   (SS*CB)     // coefs d-stride = 152
#define ICH  8           // i-planes per reduction block

// Workspace layout (float offsets)
#define OFF_SI  0                       // [n][d][64][64]  65536  (atomically accumulated)
#define OFF_SJ  65536
#define OFF_SK  131072
#define OFF_V1  196608                  // [n][d][64]      1024
#define OFF_V2  197632
#define OFF_V3  198656
#define OFF_BJK 199680                  // [n][s][64][64]  65536
#define OFF_BIK 265216
#define OFF_BIJ 330752
#define OFF_UI  396288                  // [n][s][64]      1024
#define OFF_UJ  397312
#define OFF_UK  398336
// total = 399360 floats = 1.52 MiB

typedef __attribute__((ext_vector_type(2))) float v2f;
typedef __attribute__((ext_vector_type(8))) float v8f;

// ---------------------------------------------------------------------------
// Stage 0: zero the Si accumulator region (graph-capture-safe, no memset API).
// ---------------------------------------------------------------------------
__global__ __launch_bounds__(256) void zero_kernel(float* __restrict__ ws) {
    const int idx = blockIdx.x * 256 + threadIdx.x;
    ws[OFF_SI + idx] = 0.0f;            // 16384 floats = 16 blocks x 256 x 4
    ws[OFF_SI + 16384 + idx] = 0.0f;
    ws[OFF_SI + 32768 + idx] = 0.0f;
    ws[OFF_SI + 49152 + idx] = 0.0f;
}

// ---------------------------------------------------------------------------
// Stage 1 (single pass over x): block = one 8-plane i-chunk of one (n,d)
// slice (16 slices x 8 chunks = 128 blocks, 256 threads).
//  - Si[j,k]: each thread owns sites q = tid + 256m (m=0..15) for EVERY i
//             -> 16 register accumulators, merged via global_atomic_add_f32.
//  - Sj[i,k]: a thread's 16 sites per i-plane share k = tid&63 -> one register
//             partial per i, folded with ds_add_f32 (4-way: tid, tid+64, ...).
//  - Sk[i,j]: within a wave, the 32 lanes cover half a k-row with uniform j
//             -> 5-step __shfl_xor wave32 tree, then one ds_add_f32 per wave.
// ---------------------------------------------------------------------------
__global__ __launch_bounds__(256) void reduce_kernel(const float* __restrict__ x,
                                                     float* __restrict__ ws) {
    __shared__ float lSj[ICH * MM];     // Sj partial for this i-chunk (2 KB)
    __shared__ float lSk[ICH * MM];     // Sk partial for this i-chunk (2 KB)

    const int blk   = blockIdx.x;
    const int slice = blk >> 3;                         // n*8 + d
    const int i0    = (blk & 7) * ICH;
    const float* xs = x + (size_t)slice * M3 + (size_t)i0 * M2;
    const int t     = threadIdx.x;
    const int lane  = t & 31;
    const int kk    = t & 63;                           // this thread's fixed k

    for (int q = t; q < ICH * MM; q += 256) { lSj[q] = 0.0f; lSk[q] = 0.0f; }
    __syncthreads();

    float regSi[16];
    #pragma unroll
    for (int m = 0; m < 16; ++m) regSi[m] = 0.0f;

    for (int il = 0; il < ICH; ++il) {
        float accJ = 0.0f;
        #pragma unroll
        for (int m = 0; m < 16; ++m) {
            const int q = t + 256 * m;                  // (j,k) site
            const float val = xs[il * M2 + q];
            regSi[m] += val;                            // Si partial (register)
            accJ     += val;                            // Sj partial (fixed k)
            // Sk: reduce the wave's 32 consecutive k's (uniform j per wave)
            float r = val;
            r += __shfl_xor(r, 1, 32);
            r += __shfl_xor(r, 2, 32);
            r += __shfl_xor(r, 4, 32);
            r += __shfl_xor(r, 8, 32);
            r += __shfl_xor(r, 16, 32);
            if (lane == 0) {
                const int j = q >> 6;                   // uniform across wave
                atomicAdd(&lSk[il * MM + j], r);
            }
        }
        atomicAdd(&lSj[il * MM + kk], accJ);            // 4-way fold over j%4
    }
    __syncthreads();

    // Sj/Sk rows for this i-chunk are exclusively owned: plain global stores.
    float* Sj = ws + OFF_SJ + (size_t)slice * M2 + (size_t)i0 * MM;
    float* Sk = ws + OFF_SK + (size_t)slice * M2 + (size_t)i0 * MM;
    for (int q = t; q < ICH * MM; q += 256) { Sj[q] = lSj[q]; Sk[q] = lSk[q]; }

    // Si: 8 chunks per slice merge through L2 float atomics.
    float* Si = ws + OFF_SI + (size_t)slice * M2;
    #pragma unroll
    for (int m = 0; m < 16; ++m) atomicAdd(&Si[t + 256 * m], regSi[m]);
}

// ---------------------------------------------------------------------------
// Stage 1b: double-sum vectors from the completed matrices.
//   V1[k] = sum_j Si[j,k]  (= sum over i,j)
//   V2[j] = sum_i Sk[i,j]  (= sum over i,k)
//   V3[i] = sum_j Sk[i,j]  (= sum over j,k)
// 16 blocks (one per slice), 64 threads.
// ---------------------------------------------------------------------------
__global__ __launch_bounds__(64) void vec_kernel(float* __restrict__ ws) {
    const int slice = blockIdx.x;
    const int q = threadIdx.x;
    const float* Si = ws + OFF_SI + (size_t)slice * M2;
    const float* Sk = ws + OFF_SK + (size_t)slice * M2;
    float v1 = 0.0f, v2 = 0.0f, v3 = 0.0f;
    for (int r = 0; r < MM; ++r) {
        v1 += Si[r * MM + q];
        v2 += Sk[r * MM + q];
        v3 += Sk[q * MM + r];
    }
    ws[OFF_V1 + slice * MM + q] = v1;
    ws[OFF_V2 + slice * MM + q] = v2;
    ws[OFF_V3 + slice * MM + q] = v3;
}

// ---------------------------------------------------------------------------
// Stage 2: mix reduced tensors over d with coefs -> per (n,s):
//   Bjk[j,k] = sum_d c1*Si + c4*Sj + c7*Sk      (terms broadcast along i)
//   Bik[i,k] = sum_d c2*Si + c5*Sj + c8*Sk      (broadcast along j)
//   Bij[i,j] = sum_d c3*Si + c6*Sj + c9*Sk      (broadcast along k)
//   u_i[i]   = sum_d c10*V1 + c13*V2 + c16*V3   (u_j: 11/14/17, u_k: 12/15/18)
// ---------------------------------------------------------------------------
__global__ __launch_bounds__(256) void mix_kernel(const float* __restrict__ coefs,
                                                  float* __restrict__ ws) {
    const int ns = blockIdx.x;          // n*8 + s
    const int n  = ns >> 3;
    const int s  = ns & 7;

    for (int q = threadIdx.x; q < M2; q += 256) {
        float bjk = 0.0f, bik = 0.0f, bij = 0.0f;
        #pragma unroll
        for (int d = 0; d < DD; ++d) {
            const float* c = coefs + d * CD + s * CB;
            const int sd = (n * DD + d) * M2 + q;
            const float si = ws[OFF_SI + sd];
            const float sj = ws[OFF_SJ + sd];
            const float sk = ws[OFF_SK + sd];
            bjk += c[1] * si + c[4] * sj + c[7] * sk;
            bik += c[2] * si + c[5] * sj + c[8] * sk;
            bij += c[3] * si + c[6] * sj + c[9] * sk;
        }
        ws[OFF_BJK + ns * M2 + q] = bjk;
        ws[OFF_BIK + ns * M2 + q] = bik;
        ws[OFF_BIJ + ns * M2 + q] = bij;
    }

    if (threadIdx.x < MM) {
        const int q = threadIdx.x;
        float ui = 0.0f, uj = 0.0f, uk = 0.0f;
        #pragma unroll
        for (int d = 0; d < DD; ++d) {
            const float* c = coefs + d * CD + s * CB;
            const int vd = (n * DD + d) * MM + q;
            const float v1 = ws[OFF_V1 + vd];
            const float v2 = ws[OFF_V2 + vd];
            const float v3 = ws[OFF_V3 + vd];
            ui += c[10] * v1 + c[13] * v2 + c[16] * v3;
            uj += c[11] * v1 + c[14] * v2 + c[17] * v3;
            uk += c[12] * v1 + c[15] * v2 + c[18] * v3;
        }
        ws[OFF_UI + ns * MM + q] = ui;
        ws[OFF_UJ + ns * MM + q] = uj;
        ws[OFF_UK + ns * MM + q] = uk;
    }
}

// ---------------------------------------------------------------------------
// Stage 3 (bandwidth-dominant): identity-term channel mix via
// V_WMMA_F32_16X16X4_F32 + broadcast adds + store.
// Per wave: one tile of 16 consecutive spatial sites p (same i,j; k varies).
//   A (16x4, M=s, K=d): lane<16 holds K=0,1 in v0,v1; lanes 16-31 hold K=2,3.
//   B (4x16, K=d, N=p): same per-lane K split, N = lane&15.
//   Two WMMAs accumulate d=0..3 and d=4..7.
//   D (16x16 f32): VGPR r, lanes 0-15 -> (s=r, p=p0+lane); rows 8-15 unused.
// ---------------------------------------------------------------------------
__global__ __launch_bounds__(256) void wmma_out_kernel(const float* __restrict__ x,
                                                       const float* __restrict__ coefs,
                                                       const float* __restrict__ bias,
                                                       const float* __restrict__ ws,
                                                       float* __restrict__ out) {
    const int lane  = threadIdx.x & 31;
    const int wave  = threadIdx.x >> 5;
    const int gw    = blockIdx.x * 8 + wave;
    const int n     = gw >> 14;          // 16384 tiles per n
    const int tile  = gw & 16383;
    const int p0    = tile << 4;
    const int col   = lane & 15;         // M (=s) for A, N (=p offset) for B
    const int kh    = lane >> 4;         // K half select: 0 -> d{0,1}, 1 -> d{2,3}

    // A: coefs[:, :, 0], zero rows for s >= 8 (branch-free mask).
    const float msk = (col < SS) ? 1.0f : 0.0f;
    const int   sidx = col & 7;
    v2f a0, a1;
    a0.x = msk * coefs[(2 * kh + 0) * CD + sidx * CB];
    a0.y = msk * coefs[(2 * kh + 1) * CD + sidx * CB];
    a1.x = msk * coefs[(4 + 2 * kh) * CD + sidx * CB];
    a1.y = msk * coefs[(5 + 2 * kh) * CD + sidx * CB];

    // B: x[d, p0+col] for this lane's K pair.
    const size_t xb = (size_t)n * DD * M3 + (size_t)p0 + col;
    v2f b0, b1;
    b0.x = x[xb + (size_t)(2 * kh + 0) * M3];
    b0.y = x[xb + (size_t)(2 * kh + 1) * M3];
    b1.x = x[xb + (size_t)(4 + 2 * kh) * M3];
    b1.y = x[xb + (size_t)(5 + 2 * kh) * M3];

    v8f acc = {};
    acc = __builtin_amdgcn_wmma_f32_16x16x4_f32(false, a0, false, b0,
                                                (short)0, acc, false, false);
    acc = __builtin_amdgcn_wmma_f32_16x16x4_f32(false, a1, false, b1,
                                                (short)0, acc, false, false);

    // Broadcast terms + store (lanes 0-15 hold the valid s=0..7 rows).
    if (lane < 16) {
        const int p = p0 + lane;
        const int k = p & 63;
        const int j = (p >> 6) & 63;
        const int i = p >> 12;
        #pragma unroll
        for (int r = 0; r < SS; ++r) {
            const int ns = n * SS + r;
            const float v = acc[r]
                + ws[OFF_BJK + ns * M2 + j * MM + k]
                + ws[OFF_BIK + ns * M2 + i * MM + k]
                + ws[OFF_BIJ + ns * M2 + i * MM + j]
                + ws[OFF_UI + ns * MM + i]
                + ws[OFF_UJ + ns * MM + j]
                + ws[OFF_UK + ns * MM + k]
                + bias[r];
            out[(size_t)ns * M3 + p] = v;
        }
    }
}

extern "C" void kernel_launch(void* const* d_in, const int* in_sizes, int n_in,
                              void* d_out, int out_size, void* d_ws, size_t ws_size,
                              hipStream_t stream) {
    const float* x     = (const float*)d_in[0];   // (2,8,64,64,64) f32
    const float* coefs = (const float*)d_in[1];   // (8,8,19) f32
    const float* bias  = (const float*)d_in[2];   // (1,8,1,1,1) f32
    float* out = (float*)d_out;                   // (2,8,64,64,64) f32
    float* ws  = (float*)d_ws;

    zero_kernel<<<16, 256, 0, stream>>>(ws);                       // Si := 0
    reduce_kernel<<<NN * DD * (MM / ICH), 256, 0, stream>>>(x, ws);
    vec_kernel<<<NN * DD, 64, 0, stream>>>(ws);
    mix_kernel<<<NN * SS, 256, 0, stream>>>(coefs, ws);
    wmma_out_kernel<<<(NN * (M3 / 16)) / 8, 256, 0, stream>>>(x, coefs, bias, ws, out);
}